// CoAttention_51024211476718
// MI455X (gfx1250) — compile-verified
//
#include <hip/hip_runtime.h>
#include <hip/hip_bf16.h>
#include <math.h>

typedef float v2f __attribute__((ext_vector_type(2)));
typedef float v8f __attribute__((ext_vector_type(8)));

#define D 64

// D = A(16x4) * B(4x16) + C, exact fp32 via V_WMMA_F32_16X16X4_F32
__device__ __forceinline__ v8f wmma_f32_16x16x4(v2f a, v2f b, v8f c) {
    return __builtin_amdgcn_wmma_f32_16x16x4_f32(
        /*neg_a=*/false, a, /*neg_b=*/false, b,
        /*c_mod=*/(short)0, c, /*reuse_a=*/false, /*reuse_b=*/false);
}

// ---------------------------------------------------------------------------
// Fill kernel (cannot hipMemsetAsync reliably inside capture; use a kernel)
// ---------------------------------------------------------------------------
__global__ void fill_kernel(float* __restrict__ p, float val, int n) {
    int i = blockIdx.x * blockDim.x + threadIdx.x;
    if (i < n) p[i] = val;
}

// ---------------------------------------------------------------------------
// K = X @ Wk, V = X @ Wv.  One wave -> one 16x16 output tile.
// Block = 128 threads = 4 waves = the 4 column tiles of D=64.
// A 16x4 f32 frag: lane L: m = L&15, holds K = (L>>4)*2 + {0,1}.
// B 4x16 f32 frag: lane L: n = L&15, VGPR v holds row K = (L>>4)*2 + v.
// C 16x16 f32: VGPR v, lane L: M = v + (L>>4)*8, N = L&15.
// ---------------------------------------------------------------------------
__global__ void node_proj_kernel(const float* __restrict__ X,
                                 const float* __restrict__ Wk,
                                 const float* __restrict__ Wv,
                                 float* __restrict__ K,
                                 float* __restrict__ V,
                                 int n_rows) {
    const int wave = threadIdx.x >> 5;
    const int lane = threadIdx.x & 31;
    const int m0   = blockIdx.x * 16;
    if (m0 >= n_rows) return;
    const int n0   = wave * 16;
    const int m    = lane & 15;      // also the N index for B/C frags
    const int half = lane >> 4;
    const int row  = m0 + m;

    v8f ck = {};
    v8f cv = {};
#pragma unroll
    for (int kk = 0; kk < D / 4; ++kk) {
        const int k0 = kk * 4;
        v2f a;
        a.x = X[row * D + k0 + half * 2 + 0];
        a.y = X[row * D + k0 + half * 2 + 1];
        v2f bk, bv;
        bk.x = Wk[(k0 + half * 2 + 0) * D + n0 + m];
        bk.y = Wk[(k0 + half * 2 + 1) * D + n0 + m];
        bv.x = Wv[(k0 + half * 2 + 0) * D + n0 + m];
        bv.y = Wv[(k0 + half * 2 + 1) * D + n0 + m];
        ck = wmma_f32_16x16x4(a, bk, ck);
        cv = wmma_f32_16x16x4(a, bv, cv);
    }
#pragma unroll
    for (int v = 0; v < 8; ++v) {
        const int mo = m0 + v + half * 8;
        K[mo * D + n0 + m] = ck[v];
        V[mo * D + n0 + m] = cv[v];
    }
}

// ---------------------------------------------------------------------------
// out = leaky_relu(A @ Wo + bo)
// ---------------------------------------------------------------------------
__global__ void out_proj_kernel(const float* __restrict__ A,
                                const float* __restrict__ Wo,
                                const float* __restrict__ bo,
                                float* __restrict__ out,
                                int n_rows) {
    const int wave = threadIdx.x >> 5;
    const int lane = threadIdx.x & 31;
    const int m0   = blockIdx.x * 16;
    if (m0 >= n_rows) return;
    const int n0   = wave * 16;
    const int m    = lane & 15;
    const int half = lane >> 4;
    const int row  = m0 + m;

    v8f c = {};
#pragma unroll
    for (int kk = 0; kk < D / 4; ++kk) {
        const int k0 = kk * 4;
        v2f a;
        a.x = A[row * D + k0 + half * 2 + 0];
        a.y = A[row * D + k0 + half * 2 + 1];
        v2f b;
        b.x = Wo[(k0 + half * 2 + 0) * D + n0 + m];
        b.y = Wo[(k0 + half * 2 + 1) * D + n0 + m];
        c = wmma_f32_16x16x4(a, b, c);
    }
    const float bias = bo[n0 + m];
#pragma unroll
    for (int v = 0; v < 8; ++v) {
        const int mo  = m0 + v + half * 8;
        float val = c[v] + bias;
        out[mo * D + n0 + m] = (val > 0.f) ? val : 0.01f * val;
    }
}

// ---------------------------------------------------------------------------
// translation[e] = dot(K1[i1[e]], K2[i2[e]])   (64-dim dot, float4 loads)
// ---------------------------------------------------------------------------
__global__ void edge_logit_kernel(const float* __restrict__ K1,
                                  const float* __restrict__ K2,
                                  const int* __restrict__ i1,
                                  const int* __restrict__ i2,
                                  float* __restrict__ trans, int E) {
    int e = blockIdx.x * blockDim.x + threadIdx.x;
    if (e >= E) return;
    const float4* a = (const float4*)(K1 + (size_t)i1[e] * D);
    const float4* b = (const float4*)(K2 + (size_t)i2[e] * D);
    float s = 0.f;
#pragma unroll
    for (int q = 0; q < D / 4; ++q) {
        float4 x = a[q], y = b[q];
        s = fmaf(x.x, y.x, s);
        s = fmaf(x.y, y.y, s);
        s = fmaf(x.z, y.z, s);
        s = fmaf(x.w, y.w, s);
    }
    trans[e] = s;
}

// float atomic max via signed/unsigned int reinterpretation (init = -inf)
__device__ __forceinline__ void atomicMaxFloat(float* addr, float val) {
    if (val >= 0.f) atomicMax((int*)addr, __float_as_int(val));
    else            atomicMin((unsigned int*)addr, __float_as_uint(val));
}

__global__ void edge_max_kernel(const float* __restrict__ trans,
                                const int* __restrict__ i1,
                                const int* __restrict__ i2,
                                float* __restrict__ max1,
                                float* __restrict__ max2, int E) {
    int e = blockIdx.x * blockDim.x + threadIdx.x;
    if (e >= E) return;
    float t = trans[e];
    atomicMaxFloat(&max1[i1[e]], t);
    atomicMaxFloat(&max2[i2[e]], t);
}

// e = exp((t - max)/temperature); store unnormalized into d_out edge slots,
// accumulate segment sums.
__global__ void edge_expsum_kernel(const float* __restrict__ trans,
                                   const int* __restrict__ i1,
                                   const int* __restrict__ i2,
                                   const float* __restrict__ max1,
                                   const float* __restrict__ max2,
                                   float* __restrict__ sum1,
                                   float* __restrict__ sum2,
                                   float* __restrict__ edge1,
                                   float* __restrict__ edge2,
                                   float invTemp, int E) {
    int e = blockIdx.x * blockDim.x + threadIdx.x;
    if (e >= E) return;
    int a = i1[e], b = i2[e];
    float t  = trans[e];
    float e1 = expf((t - max1[a]) * invTemp);
    float e2 = expf((t - max2[b]) * invTemp);
    edge1[e] = e1;
    edge2[e] = e2;
    atomicAdd(&sum1[a], e1);
    atomicAdd(&sum2[b], e2);
}

__global__ void edge_norm_kernel(const int* __restrict__ i1,
                                 const int* __restrict__ i2,
                                 const float* __restrict__ sum1,
                                 const float* __restrict__ sum2,
                                 float* __restrict__ edge1,
                                 float* __restrict__ edge2, int E) {
    int e = blockIdx.x * blockDim.x + threadIdx.x;
    if (e >= E) return;
    edge1[e] = edge1[e] / (sum1[i1[e]] + 1e-8f);
    edge2[e] = edge2[e] / (sum2[i2[e]] + 1e-8f);
}

// acc1[i1[e]] += w1[e] * V2[i2[e]] ; acc2[i2[e]] += w2[e] * V1[i1[e]]
// 64 threads (features) per edge.
__global__ void scatter_kernel(const float* __restrict__ w1,
                               const float* __restrict__ w2,
                               const int* __restrict__ i1,
                               const int* __restrict__ i2,
                               const float* __restrict__ V1,
                               const float* __restrict__ V2,
                               float* __restrict__ acc1,
                               float* __restrict__ acc2, int E) {
    int idx = blockIdx.x * blockDim.x + threadIdx.x;
    int e = idx >> 6;
    int f = idx & 63;
    if (e >= E) return;
    int a = i1[e], b = i2[e];
    atomicAdd(&acc1[a * D + f], w1[e] * V2[b * D + f]);
    atomicAdd(&acc2[b * D + f], w2[e] * V1[a * D + f]);
}

// ---------------------------------------------------------------------------
extern "C" void kernel_launch(void* const* d_in, const int* in_sizes, int n_in,
                              void* d_out, int out_size, void* d_ws, size_t ws_size,
                              hipStream_t stream) {
    const float* node1  = (const float*)d_in[0];
    const int*   seg_i1 = (const int*)d_in[1];
    const float* node2  = (const float*)d_in[3];
    const int*   seg_i2 = (const int*)d_in[4];
    const float* Wk     = (const float*)d_in[6];
    const float* Wv     = (const float*)d_in[7];
    const float* Wo     = (const float*)d_in[8];
    const float* bo     = (const float*)d_in[9];

    const int N1 = in_sizes[0] / D;
    const int N2 = in_sizes[3] / D;
    const int E  = in_sizes[1];

    // workspace layout (floats)
    float* ws   = (float*)d_ws;
    float* K1   = ws;                  ws += (size_t)N1 * D;
    float* K2   = ws;                  ws += (size_t)N2 * D;
    float* V1   = ws;                  ws += (size_t)N1 * D;
    float* V2   = ws;                  ws += (size_t)N2 * D;
    float* trans= ws;                  ws += (size_t)E;
    float* acc1 = ws;                  ws += (size_t)N1 * D;   // zero block start
    float* acc2 = ws;                  ws += (size_t)N2 * D;
    float* sum1 = ws;                  ws += (size_t)N1;
    float* sum2 = ws;                  ws += (size_t)N2;       // zero block end
    float* max1 = ws;                  ws += (size_t)N1;       // -inf block
    float* max2 = ws;                  ws += (size_t)N2;

    // output layout: msg1, msg2, edge1, edge2
    float* out   = (float*)d_out;
    float* msg1  = out;
    float* msg2  = out + (size_t)N1 * D;
    float* edge1 = out + (size_t)(N1 + N2) * D;
    float* edge2 = edge1 + (size_t)E;

    const int TB = 256;

    // init accumulators / sums to 0, maxes to -inf
    const int nzero = N1 * D + N2 * D + N1 + N2;
    fill_kernel<<<(nzero + TB - 1) / TB, TB, 0, stream>>>(acc1, 0.0f, nzero);
    const int ninf = N1 + N2;
    fill_kernel<<<(ninf + TB - 1) / TB, TB, 0, stream>>>(max1, -INFINITY, ninf);

    // K/V projections (fp32 WMMA)
    node_proj_kernel<<<(N1 + 15) / 16, 128, 0, stream>>>(node1, Wk, Wv, K1, V1, N1);
    node_proj_kernel<<<(N2 + 15) / 16, 128, 0, stream>>>(node2, Wk, Wv, K2, V2, N2);

    const int egrid = (E + TB - 1) / TB;
    edge_logit_kernel<<<egrid, TB, 0, stream>>>(K1, K2, seg_i1, seg_i2, trans, E);
    edge_max_kernel<<<egrid, TB, 0, stream>>>(trans, seg_i1, seg_i2, max1, max2, E);
    edge_expsum_kernel<<<egrid, TB, 0, stream>>>(trans, seg_i1, seg_i2, max1, max2,
                                                 sum1, sum2, edge1, edge2,
                                                 0.125f /* 1/sqrt(64) */, E);
    edge_norm_kernel<<<egrid, TB, 0, stream>>>(seg_i1, seg_i2, sum1, sum2,
                                               edge1, edge2, E);

    const long long stot = (long long)E * D;
    scatter_kernel<<<(int)((stot + TB - 1) / TB), TB, 0, stream>>>(
        edge1, edge2, seg_i1, seg_i2, V1, V2, acc1, acc2, E);

    // output projections with fused bias + leaky relu (fp32 WMMA)
    out_proj_kernel<<<(N1 + 15) / 16, 128, 0, stream>>>(acc1, Wo, bo, msg1, N1);
    out_proj_kernel<<<(N2 + 15) / 16, 128, 0, stream>>>(acc2, Wo, bo, msg2, N2);
}